// ZBLRepulsion_57062935495397
// MI455X (gfx1250) — compile-verified
//
#include <hip/hip_runtime.h>
#include <stdint.h>

#define TPB   256
#define GRID  512
#define CHUNK 2048          // edges per TDM tile; 2 streams x 2 buffers = 32KB LDS

#define NPAR  10            // precomputed params
#define NZTAB 64            // Z^ae table entries

typedef unsigned int u32x4 __attribute__((ext_vector_type(4)));
typedef int          i32x4 __attribute__((ext_vector_type(4)));
typedef int          i32x8 __attribute__((ext_vector_type(8)));

#if __has_builtin(__builtin_amdgcn_tensor_load_to_lds)
#define HAVE_TDM 1
#else
#define HAVE_TDM 0
#endif

__device__ __forceinline__ float softplus_f(float x) {
    return log1pf(expf(x));
}

#if HAVE_TDM
// 1D TDM DMA: n_valid elements of 4B data from gptr -> LDS byte address lds_addr.
// tile_dim0 = tile0; reads past tensor_dim0 return zero (harmless: idx 0/0 is a
// masked self-pair, and the compute loop is bounds-guarded anyway).
__device__ __forceinline__ void tdm_load_1d(uint32_t lds_addr, const void* gptr,
                                            int n_valid, int tile0) {
    uint64_t ga = (uint64_t)(uintptr_t)gptr;
    u32x4 g0;
    g0[0] = 1u;                                   // count=1, user descriptor
    g0[1] = lds_addr;                             // lds_addr [63:32]
    g0[2] = (uint32_t)ga;                         // global_addr [95:64]
    g0[3] = (uint32_t)((ga >> 32) & 0x01FFFFFFull) | (2u << 30); // addr[120:96], type=2
    i32x8 g1;
    g1[0] = (int)(2u << 16);                      // wg_mask=0, data_size=2 (4B)
    g1[1] = (int)(((uint32_t)n_valid & 0xFFFFu) << 16);                // tensor_dim0 lo16 @bit48
    g1[2] = (int)((((uint32_t)n_valid >> 16) & 0xFFFFu) | (1u << 16)); // dim0 hi16, tensor_dim1=1
    g1[3] = (int)((uint32_t)tile0 << 16);         // tensor_dim1 hi=0, tile_dim0
    g1[4] = 1;                                    // tile_dim1=1, tile_dim2=0
    g1[5] = n_valid;                              // tensor_dim0_stride (lo)
    g1[6] = 0;
    g1[7] = 0;
    i32x4 z4 = {0, 0, 0, 0};
#if __clang_major__ >= 23
    i32x8 z8 = {0, 0, 0, 0, 0, 0, 0, 0};
    __builtin_amdgcn_tensor_load_to_lds(g0, g1, z4, z4, z8, 0);
#else
    __builtin_amdgcn_tensor_load_to_lds(g0, g1, z4, z4, 0);
#endif
}
#endif

// One tiny block: softplus all scalar params + build the Z^ae table once,
// keeping every libm inline OUT of the persistent edge kernel.
// par[0] = 1/softplus(a_num)
// par[1..4] = softplus(coefficients[k])
// par[5..8] = -softplus(exponents[k]) * log2(e)
// par[9] = 0.5 * KE * softplus(rep_scale)
// zpow[z] = z^softplus(a_exp), zpow[0] = 0
__global__ __launch_bounds__(64) void zbl_prep(const float* __restrict__ a_exp,
                                               const float* __restrict__ a_num,
                                               const float* __restrict__ coeffs,
                                               const float* __restrict__ exps,
                                               const float* __restrict__ rep_scale,
                                               float* __restrict__ par,
                                               float* __restrict__ zpow) {
    const int t = threadIdx.x;
    const float ae = softplus_f(a_exp[0]);
    if (t < NZTAB)
        zpow[t] = (t >= 1) ? powf((float)t, ae) : 0.0f;
    if (t == 0) {
        const float L2E = 1.4426950408889634f;   // exp(-e*d) = exp2(-e*L2E*d)
        par[0] = 1.0f / softplus_f(a_num[0]);
        par[1] = softplus_f(coeffs[0]);
        par[2] = softplus_f(coeffs[1]);
        par[3] = softplus_f(coeffs[2]);
        par[4] = softplus_f(coeffs[3]);
        par[5] = -softplus_f(exps[0]) * L2E;
        par[6] = -softplus_f(exps[1]) * L2E;
        par[7] = -softplus_f(exps[2]) * L2E;
        par[8] = -softplus_f(exps[3]) * L2E;
        par[9] = 0.5f * 14.3996f * softplus_f(rep_scale[0]);
    }
}

// Pack R (N,3) + Z (N) into float4 {x,y,z,(float)Z}: per-edge gather becomes a
// single b128 load per atom (working set 1.6MB -> L2/WGP$-resident).
__global__ __launch_bounds__(TPB) void zbl_pack(const float* __restrict__ R,
                                                const int* __restrict__ Z,
                                                float4* __restrict__ P, int N) {
    int i = blockIdx.x * TPB + threadIdx.x;
    if (i < N) {
        float4 p;
        p.x = R[3 * i + 0];
        p.y = R[3 * i + 1];
        p.z = R[3 * i + 2];
        p.w = (float)Z[i];
        P[i] = p;
    }
}

template <bool PACKED>
__global__ __launch_bounds__(TPB) void zbl_edges(
    const float4* __restrict__ P,
    const float*  __restrict__ R,
    const int*    __restrict__ Z,
    const int*    __restrict__ idx,       // [2*E]: idx_i then idx_j
    const float*  __restrict__ par,       // NPAR precomputed params
    const float*  __restrict__ zpow,      // NZTAB-entry Z^ae table
    float* __restrict__ partial,
    int E, int nChunks)
{
    __shared__ int   s_idx[2][2][CHUNK];  // [buf][i/j][elem]
    __shared__ float s_zpow[NZTAB];
    __shared__ float s_wsum[TPB / 32];

    const int tid  = threadIdx.x;
    const int wid  = tid >> 5;            // wave id (wave32)
    const int lane = tid & 31;

    // Uniform param loads (scalar-load friendly), no libm here.
    const float inv_an = par[0];
    const float cf0 = par[1], cf1 = par[2], cf2 = par[3], cf3 = par[4];
    const float k0  = par[5], k1  = par[6], k2  = par[7], k3  = par[8];

    if (tid < NZTAB)
        s_zpow[tid] = zpow[tid];

    const int firstChunk = (int)blockIdx.x;

#if HAVE_TDM
    // Wave-uniform guard -> scalar branch; tensor ops ignore EXEC, so exactly
    // the wave that *reaches* the instruction issues it (wave 0 only).
    if (wid == 0 && firstChunk < nChunks) {
        int cs = firstChunk * CHUNK;
        int nv = E - cs; if (nv > CHUNK) nv = CHUNK;
        tdm_load_1d((uint32_t)(uintptr_t)&s_idx[0][0][0], idx + cs,     nv, CHUNK);
        tdm_load_1d((uint32_t)(uintptr_t)&s_idx[0][1][0], idx + E + cs, nv, CHUNK);
    }
#endif

    float acc = 0.0f;
    int buf = 0;
    for (int c = firstChunk; c < nChunks; c += (int)gridDim.x) {
        const int cs  = c * CHUNK;
        const int lim = (E - cs < CHUNK) ? (E - cs) : CHUNK;

#if HAVE_TDM
        if (wid == 0)
            __builtin_amdgcn_s_wait_tensorcnt(0);   // tile c landed in LDS
        __syncthreads();                            // visible to all waves (also covers s_zpow)
        // DMA tile c+gridDim into the other buffer; overlaps with compute below.
        int cn = c + (int)gridDim.x;
        if (wid == 0 && cn < nChunks) {
            int cs2 = cn * CHUNK;
            int nv = E - cs2; if (nv > CHUNK) nv = CHUNK;
            tdm_load_1d((uint32_t)(uintptr_t)&s_idx[buf ^ 1][0][0], idx + cs2,     nv, CHUNK);
            tdm_load_1d((uint32_t)(uintptr_t)&s_idx[buf ^ 1][1][0], idx + E + cs2, nv, CHUNK);
        }
#else
        __syncthreads();
#endif

        for (int t = tid; t < lim; t += TPB) {
#if HAVE_TDM
            const int i = s_idx[buf][0][t];
            const int j = s_idx[buf][1][t];
#else
            const int i = idx[cs + t];
            const int j = idx[E + cs + t];
#endif
            float xi, yi, zi, zfi, xj, yj, zj, zfj;
            if (PACKED) {
                float4 pi = P[i], pj = P[j];
                xi = pi.x; yi = pi.y; zi = pi.z; zfi = pi.w;
                xj = pj.x; yj = pj.y; zj = pj.z; zfj = pj.w;
            } else {
                xi = R[3 * i]; yi = R[3 * i + 1]; zi = R[3 * i + 2]; zfi = (float)Z[i];
                xj = R[3 * j]; yj = R[3 * j + 1]; zj = R[3 * j + 2]; zfj = (float)Z[j];
            }
            float dx = xj - xi, dy = yj - yi, dz = zj - zi;
            float r2 = dx * dx + dy * dy + dz * dz;
            // r2 >= 36: clamped dr=6 -> cos(pi) == -1.0f exactly -> contribution 0.
            if ((i != j) && (r2 < 36.0f)) {
                float dr = fmaxf(sqrtf(r2), 0.02f);
                float cc = 0.5f * (__cosf(dr * 0.52359877559829887f) + 1.0f); // pi/6
                float dist = dr * (s_zpow[(int)zfi] + s_zpow[(int)zfj]) * inv_an;
                float f = cf0 * exp2f(k0 * dist) + cf1 * exp2f(k1 * dist)
                        + cf2 * exp2f(k2 * dist) + cf3 * exp2f(k3 * dist);
                acc += zfi * zfj * f * cc / dr;
            }
        }
        __syncthreads();   // done reading buf before it is re-DMAed (next iter)
        buf ^= 1;
    }

    // wave32 reduction, then cross-wave via LDS — deterministic (no atomics).
    for (int off = 16; off > 0; off >>= 1)
        acc += __shfl_down(acc, off, 32);
    if (lane == 0) s_wsum[wid] = acc;
    __syncthreads();
    if (wid == 0) {
        float v = (lane < TPB / 32) ? s_wsum[lane] : 0.0f;
        for (int off = 4; off > 0; off >>= 1)
            v += __shfl_down(v, off, 32);
        if (lane == 0) partial[blockIdx.x] = v;
    }
}

__global__ __launch_bounds__(TPB) void zbl_reduce(const float* __restrict__ partial, int n,
                                                  const float* __restrict__ par,
                                                  float* __restrict__ out) {
    __shared__ float s[TPB / 32];
    int tid = threadIdx.x;
    float acc = 0.0f;
    for (int i = tid; i < n; i += TPB) acc += partial[i];
    for (int off = 16; off > 0; off >>= 1) acc += __shfl_down(acc, off, 32);
    if ((tid & 31) == 0) s[tid >> 5] = acc;
    __syncthreads();
    if (tid < 32) {
        float v = (tid < TPB / 32) ? s[tid] : 0.0f;
        for (int off = 4; off > 0; off >>= 1) v += __shfl_down(v, off, 32);
        if (tid == 0)
            out[0] = par[9] * v;   // 0.5 * KE * softplus(rep_scale) * sum
    }
}

extern "C" void kernel_launch(void* const* d_in, const int* in_sizes, int n_in,
                              void* d_out, int out_size, void* d_ws, size_t ws_size,
                              hipStream_t stream) {
    const float* R         = (const float*)d_in[0];
    const int*   Z         = (const int*)  d_in[1];
    const int*   idx       = (const int*)  d_in[2];
    const float* a_exp     = (const float*)d_in[3];
    const float* a_num     = (const float*)d_in[4];
    const float* coeffs    = (const float*)d_in[5];
    const float* exps      = (const float*)d_in[6];
    const float* rep_scale = (const float*)d_in[7];

    const int N = in_sizes[0] / 3;
    const int E = in_sizes[2] / 2;
    const int nChunks = (E + CHUNK - 1) / CHUNK;

    const size_t packBytes  = (size_t)N * sizeof(float4);
    const size_t smallBytes = (size_t)(GRID + NPAR + NZTAB + 16) * sizeof(float);
    const bool packed = ws_size >= packBytes + smallBytes;

    // ws layout: [P (if packed)] [partial: GRID] [par: NPAR] [zpow: NZTAB]
    char* w = (char*)d_ws;
    float4* P = nullptr;
    if (packed) { P = (float4*)w; w += packBytes; }
    float* partial = (float*)w; w += (size_t)GRID * sizeof(float);
    float* par     = (float*)w; w += (size_t)NPAR * sizeof(float);
    float* zpow    = (float*)w;

    zbl_prep<<<1, 64, 0, stream>>>(a_exp, a_num, coeffs, exps, rep_scale, par, zpow);
    if (packed) {
        zbl_pack<<<(N + TPB - 1) / TPB, TPB, 0, stream>>>(R, Z, P, N);
        zbl_edges<true><<<GRID, TPB, 0, stream>>>(P, R, Z, idx, par, zpow,
                                                  partial, E, nChunks);
    } else {
        zbl_edges<false><<<GRID, TPB, 0, stream>>>(P, R, Z, idx, par, zpow,
                                                   partial, E, nChunks);
    }
    zbl_reduce<<<1, TPB, 0, stream>>>(partial, GRID, par, (float*)d_out);
}